// GCNClassifier_14774687498490
// MI455X (gfx1250) — compile-verified
//
#include <hip/hip_runtime.h>
#include <hip/hip_bf16.h>

#define N_NODES 50000
#define N_EDGES 1600000
#define D 128
#define N_GRAPHS 64
#define N_CLASSES 8

typedef __attribute__((ext_vector_type(2))) float v2f;
typedef __attribute__((ext_vector_type(8))) float v8f;

static inline int cdiv(long a, long b) { return (int)((a + b - 1) / b); }

// ---------------- degree / norm ----------------

__global__ __launch_bounds__(256) void deg_kernel(const int* __restrict__ dst,
                                                  float* __restrict__ deg, int E) {
    int e = blockIdx.x * 256 + threadIdx.x;
    if (e < E) atomicAdd(&deg[dst[e]], 1.0f);
}

__global__ __launch_bounds__(256) void invsqrt_kernel(float* __restrict__ deg, int n) {
    int i = blockIdx.x * 256 + threadIdx.x;
    if (i < n) {
        float dg = deg[i];
        deg[i] = (dg > 0.0f) ? rsqrtf(fmaxf(dg, 1.0f)) : 0.0f;
    }
}

__global__ __launch_bounds__(256) void norm_kernel(const int* __restrict__ src,
                                                   const int* __restrict__ dst,
                                                   const float* __restrict__ dis,
                                                   float* __restrict__ norm, int E) {
    int e = blockIdx.x * 256 + threadIdx.x;
    if (e < E) norm[e] = dis[src[e]] * dis[dst[e]];
}

// ---------------- edge scatter-aggregate: agg[dst] += h[src] * norm ----------------
// one thread handles one edge x 4 channels (float4 gather, 4 atomic f32 adds)

__global__ __launch_bounds__(256) void scatter_kernel(const float* __restrict__ h,
                                                      const int* __restrict__ src,
                                                      const int* __restrict__ dst,
                                                      const float* __restrict__ norm,
                                                      float* __restrict__ agg) {
    long t = (long)blockIdx.x * 256 + threadIdx.x;  // N_EDGES * 32 threads
    if (t >= (long)N_EDGES * 32) return;
    int e  = (int)(t >> 5);
    int c4 = (int)(t & 31) << 2;
    int s = src[e], d = dst[e];
    float nv = norm[e];
    float4 xv = *reinterpret_cast<const float4*>(h + (long)s * D + c4);
    float* ap = agg + (long)d * D + c4;
    atomicAdd(ap + 0, xv.x * nv);
    atomicAdd(ap + 1, xv.y * nv);
    atomicAdd(ap + 2, xv.z * nv);
    atomicAdd(ap + 3, xv.w * nv);
}

// ---------------- fp32 WMMA GEMM: out = relu(A[M,128] @ W[128,128] + bias) ----------------
// one wave computes one 16x16 tile; K-loop of 32 x v_wmma_f32_16x16x4_f32

__global__ __launch_bounds__(256) void gemm_bias_relu(const float* __restrict__ A,
                                                      const float* __restrict__ W,
                                                      const float* __restrict__ bias,
                                                      float* __restrict__ out, int M) {
    const int lane = threadIdx.x & 31;
    const int wave = threadIdx.x >> 5;
    const int waveId = blockIdx.x * 8 + wave;
    const int tilesN = D / 16;                 // 8
    const int m0 = (waveId / tilesN) * 16;
    const int n0 = (waveId % tilesN) * 16;
    if (m0 >= M) return;

    const int l15 = lane & 15;
    const int hi  = lane >> 4;                 // 0 or 1 (lane half)

    v8f c = {};
    const float* arow = A + (long)(m0 + l15) * D + 2 * hi;  // A[m0+l15][2*hi + ...]
    const float* bbase = W + n0 + l15;

#pragma unroll 8
    for (int k = 0; k < D; k += 4) {
        // A 16x4 f32: lane l holds K = k + 2*hi + {0,1}
        v2f a;
        a.x = arow[k + 0];
        a.y = arow[k + 1];
        // B 4x16 f32: lane l, comp j holds B[k + 2*hi + j][n0 + l15]
        v2f b;
        b.x = bbase[(long)(k + 2 * hi + 0) * D];
        b.y = bbase[(long)(k + 2 * hi + 1) * D];
        c = __builtin_amdgcn_wmma_f32_16x16x4_f32(
            /*neg_a=*/false, a, /*neg_b=*/false, b,
            /*c_mod=*/(short)0, c, /*reuse_a=*/false, /*reuse_b=*/false);
    }

    const int col = n0 + l15;
    const float bv = bias[col];
#pragma unroll
    for (int r = 0; r < 8; ++r) {
        int row = m0 + r + 8 * hi;             // C layout: VGPR r -> M = r (+8 for hi lanes)
        out[(long)row * D + col] = fmaxf(c[r] + bv, 0.0f);
    }
}

// ---------------- per-graph pooling ----------------

__global__ __launch_bounds__(256) void pool_feat_kernel(const float* __restrict__ h,
                                                        const int* __restrict__ gid,
                                                        float* __restrict__ pooled) {
    long t = (long)blockIdx.x * 256 + threadIdx.x;  // N_NODES * 32 threads
    if (t >= (long)N_NODES * 32) return;
    int n  = (int)(t >> 5);
    int c4 = (int)(t & 31) << 2;
    int g = gid[n];
    float4 hv = *reinterpret_cast<const float4*>(h + (long)n * D + c4);
    float* pp = pooled + (long)g * D + c4;
    atomicAdd(pp + 0, hv.x);
    atomicAdd(pp + 1, hv.y);
    atomicAdd(pp + 2, hv.z);
    atomicAdd(pp + 3, hv.w);
}

__global__ __launch_bounds__(256) void pool_count_kernel(const int* __restrict__ gid,
                                                         float* __restrict__ counts) {
    int n = blockIdx.x * 256 + threadIdx.x;
    if (n < N_NODES) atomicAdd(&counts[gid[n]], 1.0f);
}

// ---------------- classifier: logits = (pooled/count) @ Wc + bc ----------------

__global__ __launch_bounds__(512) void classifier_kernel(const float* __restrict__ pooled,
                                                         const float* __restrict__ counts,
                                                         const float* __restrict__ Wc,
                                                         const float* __restrict__ bc,
                                                         float* __restrict__ out) {
    int t = threadIdx.x;                        // 64*8 = 512
    int g = t >> 3;
    int cls = t & 7;
    float inv = 1.0f / fmaxf(counts[g], 1.0f);
    float s = 0.0f;
    for (int k = 0; k < D; ++k)
        s += pooled[(long)g * D + k] * Wc[(long)k * N_CLASSES + cls];
    out[(long)g * N_CLASSES + cls] = s * inv + bc[cls];
}

extern "C" void kernel_launch(void* const* d_in, const int* in_sizes, int n_in,
                              void* d_out, int out_size, void* d_ws, size_t ws_size,
                              hipStream_t stream) {
    const float* x    = (const float*)d_in[0];
    const int*   ei   = (const int*)d_in[1];      // [2, E]
    const int*   gid  = (const int*)d_in[2];
    const float* W1   = (const float*)d_in[3];
    const float* b1   = (const float*)d_in[4];
    const float* W2   = (const float*)d_in[5];
    const float* b2   = (const float*)d_in[6];
    const float* Wc   = (const float*)d_in[7];
    const float* bc   = (const float*)d_in[8];
    float* out = (float*)d_out;

    const int* src = ei;
    const int* dst = ei + N_EDGES;

    // workspace layout (floats)
    float* ws     = (float*)d_ws;
    float* deg    = ws;                                 // N_NODES  (becomes deg^-1/2)
    float* norm   = deg + N_NODES;                      // N_EDGES
    float* agg    = norm + N_EDGES;                     // N_NODES*D (reused both layers)
    float* hbuf   = agg + (size_t)N_NODES * D;          // N_NODES*D
    float* pooled = hbuf + (size_t)N_NODES * D;         // N_GRAPHS*D
    float* counts = pooled + (size_t)N_GRAPHS * D;      // N_GRAPHS

    hipMemsetAsync(deg, 0, N_NODES * sizeof(float), stream);
    hipMemsetAsync(agg, 0, (size_t)N_NODES * D * sizeof(float), stream);
    hipMemsetAsync(pooled, 0, (size_t)(N_GRAPHS * D + N_GRAPHS) * sizeof(float), stream);

    // degree + symmetric normalization
    deg_kernel<<<cdiv(N_EDGES, 256), 256, 0, stream>>>(dst, deg, N_EDGES);
    invsqrt_kernel<<<cdiv(N_NODES, 256), 256, 0, stream>>>(deg, N_NODES);
    norm_kernel<<<cdiv(N_EDGES, 256), 256, 0, stream>>>(src, dst, deg, norm, N_EDGES);

    const int scatter_blocks = cdiv((long)N_EDGES * 32, 256);
    const int gemm_blocks = (N_NODES / 16) * (D / 16) / 8;   // 3125

    // layer 1
    scatter_kernel<<<scatter_blocks, 256, 0, stream>>>(x, src, dst, norm, agg);
    gemm_bias_relu<<<gemm_blocks, 256, 0, stream>>>(agg, W1, b1, hbuf, N_NODES);

    // layer 2
    hipMemsetAsync(agg, 0, (size_t)N_NODES * D * sizeof(float), stream);
    scatter_kernel<<<scatter_blocks, 256, 0, stream>>>(hbuf, src, dst, norm, agg);
    gemm_bias_relu<<<gemm_blocks, 256, 0, stream>>>(agg, W2, b2, hbuf, N_NODES);

    // pooling + classifier
    pool_feat_kernel<<<cdiv((long)N_NODES * 32, 256), 256, 0, stream>>>(hbuf, gid, pooled);
    pool_count_kernel<<<cdiv(N_NODES, 256), 256, 0, stream>>>(gid, counts);
    classifier_kernel<<<1, 512, 0, stream>>>(pooled, counts, Wc, bc, out);
}